// CorrectingHawkes_90065464197521
// MI455X (gfx1250) — compile-verified
//
#include <hip/hip_runtime.h>
#include <hip/hip_bf16.h>

// ---------------------------------------------------------------------------
// Correcting-Hawkes log-likelihood, N = 8192 events.
// phaseA : triangular O(N^2) row sums via V_WMMA_F32_16X16X4_F32 (B = ones),
//          then per-row quartic root. Inner loop made wave-uniform (scalar
//          loop control) so WMMA always executes with EXEC == all-ones.
// phaseB : O((N+1)*N) term_3 sweep, per-block partials.
// final  : deterministic single-block reduction -> scalar output.
// ---------------------------------------------------------------------------

#define EPSF 1e-10f

typedef float v2f __attribute__((ext_vector_type(2)));
typedef float v8f __attribute__((ext_vector_type(8)));

__device__ __forceinline__ float plusf(float x) {
    // x*(x>EPS) + EPS*(x<EPS), exactly as the reference
    return x * ((x > EPSF) ? 1.f : 0.f) + EPSF * ((x < EPSF) ? 1.f : 0.f);
}

__device__ __forceinline__ float quarticf(float a, float b, float c, float d, float e) {
    float p  = (8.f * a * c - 3.f * b * b) / (8.f * a * a);
    float q  = (b * b * b - 4.f * a * b * c + 8.f * a * a * d) / (8.f * a * a * a);
    float d0 = c * c - 3.f * b * d + 12.f * a * e;
    float d1 = 2.f * c * c * c - 9.f * b * c * d + 27.f * b * b * e
             + 27.f * a * d * d - 72.f * a * c * e;
    float t1 = d1 * d1 - 4.f * d0 * d0 * d0;
    float Q  = cbrtf((sqrtf(plusf(t1)) + d1) * 0.5f);
    float t2 = (-2.f / 3.f) * p + (Q + d0 / Q) / (3.f * a);
    float S  = sqrtf(plusf(t2)) * 0.5f;
    float t3 = -4.f * S * S - 2.f * p + q / S;
    float x1 = -b / (4.f * a) - S + sqrtf(plusf(t3)) * 0.5f;
    return ((S > 0.001f) && (x1 > 0.f)) ? x1 : 0.f;
}

__device__ __forceinline__ v8f wmma_rowsum(v2f a, v8f c) {
    v2f ones = {1.f, 1.f};
    // D = A(16x4) * ones(4x16) + C  -> every column of C holds the row sum
    return __builtin_amdgcn_wmma_f32_16x16x4_f32(
        /*neg_a=*/false, a, /*neg_b=*/false, ones,
        /*c_mod=*/(short)0, c, /*reuse_a=*/false, /*reuse_b=*/false);
}

// ---------------------------------------------------------------------------
// Phase A: one 16-row tile per block; 8 waves split the j (K) range.
// ---------------------------------------------------------------------------
__global__ __launch_bounds__(256)
void hawkes_phaseA(const float* __restrict__ his_t,
                   const float* __restrict__ p_log_mu,
                   const float* __restrict__ p_log_zeta,
                   const float* __restrict__ p_c0,
                   const float* __restrict__ p_c1,
                   const float* __restrict__ p_c2,
                   int n,
                   float* __restrict__ ends,      // [n+1]
                   float* __restrict__ loglamb,   // [n]
                   float* __restrict__ rootsarr)  // [n]
{
    __shared__ float sh_t[8192];
    __shared__ float shA[8][16];
    __shared__ float shB[8][16];
    __shared__ float shC[8][16];

    const int tid  = threadIdx.x;
    const int lane = tid & 31;
    const int r0   = blockIdx.x * 16;
    // wave index is wave-uniform: pin it to an SGPR so the k0 loop below has
    // scalar loop control (s_cmp/s_cbranch) and EXEC stays all-ones for WMMA.
    const int wave = __builtin_amdgcn_readfirstlane(tid >> 5);

    for (int j = tid; j < n; j += 256) sh_t[j] = his_t[j];
    __syncthreads();

    const float zeta = __expf(*p_log_zeta);
    const float c0 = *p_c0, c1 = *p_c1, c2 = *p_c2;

    const int m     = lane & 15;
    const int khalf = lane >> 4;
    const int i     = r0 + m;
    const float ti  = sh_t[(i < n) ? i : (n - 1)];

    v8f accA = {}; v8f accB = {}; v8f accC = {};

    const int kmax = r0 + 15;  // need j <= r0+15 (scalar bound)
    for (int k0 = wave * 4; k0 <= kmax; k0 += 32) {
        float ea[2], eb[2], ec[2];
#pragma unroll
        for (int u = 0; u < 2; ++u) {
            int   j   = k0 + khalf * 2 + u;
            float tj  = sh_t[(j < n) ? j : (n - 1)];
            float dt  = fminf(fmaxf(ti - tj, 0.f), 50.f);
            float ez  = expf(-zeta * dt);
            float w   = (j <= i) ? ez : 0.f;   // tril incl. diagonal
            ea[u] = c2 * w;
            eb[u] = (c1 + 2.f * c2 * dt) * w;
            ec[u] = (c0 + dt * (c1 + c2 * dt)) * w;
        }
        v2f vA = {ea[0], ea[1]};
        v2f vB = {eb[0], eb[1]};
        v2f vC = {ec[0], ec[1]};
        accA = wmma_rowsum(vA, accA);
        accB = wmma_rowsum(vB, accB);
        accC = wmma_rowsum(vC, accC);
    }

    // C layout: VGPR v, lane 0 -> (M=v, N=0); lane 16 -> (M=v+8, N=0)
    if (lane == 0) {
#pragma unroll
        for (int v = 0; v < 8; ++v) { shA[wave][v] = accA[v]; shB[wave][v] = accB[v]; shC[wave][v] = accC[v]; }
    } else if (lane == 16) {
#pragma unroll
        for (int v = 0; v < 8; ++v) { shA[wave][8+v] = accA[v]; shB[wave][8+v] = accB[v]; shC[wave][8+v] = accC[v]; }
    }
    __syncthreads();

    if (tid == 16 && blockIdx.x == 0) ends[0] = 0.f;

    if (tid < 16) {
        const int row = tid;
        const int ii  = r0 + row;
        float A = 0.f, B = 0.f, C = 0.f;
#pragma unroll
        for (int w = 0; w < 8; ++w) { A += shA[w][row]; B += shB[w][row]; C += shC[w][row]; }

        const float mu = __expf(*p_log_mu);
        // lamb (strict tril) = C (incl. diag) - c0
        float lam = plusf(mu + (C - c0));
        loglamb[ii] = logf(lam);

        float qa = A * 0.5f;
        float qb = -A + B * 0.5f;
        float qc = A - B + C * 0.5f;
        float qd = B - C;
        float qe = C + mu;
        float root = quarticf(qa, qb, qc, qd, qe);
        rootsarr[ii] = root;
        ends[ii + 1] = root + sh_t[ii];
    }
}

// ---------------------------------------------------------------------------
// Phase B: term_3 sweep. One block per i in [0, n]; threads stride over j.
// ---------------------------------------------------------------------------
__global__ __launch_bounds__(256)
void hawkes_term3(const float* __restrict__ his_t,
                  const float* __restrict__ p_log_zeta,
                  const float* __restrict__ p_c0,
                  const float* __restrict__ p_c1,
                  const float* __restrict__ p_c2,
                  const int* __restrict__ p_t_end,
                  int n,
                  const float* __restrict__ ends,   // [n+1]
                  float* __restrict__ partials)     // [n+1]
{
    __shared__ float sh_t[8192];
    __shared__ float red[256];

    const int tid = threadIdx.x;
    const int i   = blockIdx.x;

    for (int j = tid; j < n; j += 256) sh_t[j] = his_t[j];
    __syncthreads();

    const float zeta = __expf(*p_log_zeta);
    const float c0 = *p_c0, c1 = *p_c1, c2 = *p_c2;
    const float kc = zeta * (c1 + c0 * zeta);

    const float start_i = (i < n) ? sh_t[i] : (float)(*p_t_end);
    const float end_i   = ends[i];

    float acc = 0.f;
    for (int j = tid; j < n; j += 256) {
        float tj = sh_t[j];
        float xs = fmaxf(start_i - tj, 0.f);
        float xe = fmaxf(end_i   - tj, 0.f);

        float zs = zeta * xs;
        float Fs = expf(-zs) * (-c2 * (zs * (zs + 2.f) + 2.f) - (c1 * zs * zeta + kc));
        float ze = zeta * xe;
        float Fe = expf(-ze) * (-c2 * (ze * (ze + 2.f) + 2.f) - (c1 * ze * zeta + kc));
        acc += Fs - Fe;
    }

    red[tid] = acc;
    __syncthreads();
#pragma unroll
    for (int s = 128; s > 0; s >>= 1) {
        if (tid < s) red[tid] += red[tid + s];
        __syncthreads();
    }
    if (tid == 0) partials[i] = red[0];
}

// ---------------------------------------------------------------------------
// Finalize: deterministic single-block reduction of the three arrays.
// ---------------------------------------------------------------------------
__global__ __launch_bounds__(256)
void hawkes_final(const float* __restrict__ p_log_mu,
                  const float* __restrict__ p_log_zeta,
                  const int* __restrict__ p_t_start,
                  const int* __restrict__ p_t_end,
                  int n,
                  const float* __restrict__ loglamb,   // [n]
                  const float* __restrict__ rootsarr,  // [n]
                  const float* __restrict__ partials,  // [n+1]
                  float* __restrict__ out)
{
    __shared__ float red[256];
    const int tid = threadIdx.x;

    float s1 = 0.f, s2 = 0.f, s3 = 0.f;
    for (int k = tid; k < n; k += 256) { s1 += loglamb[k]; s2 += rootsarr[k]; }
    for (int k = tid; k < n + 1; k += 256) s3 += partials[k];

    float tot1, tot2, tot3;
    red[tid] = s1; __syncthreads();
#pragma unroll
    for (int s = 128; s > 0; s >>= 1) { if (tid < s) red[tid] += red[tid + s]; __syncthreads(); }
    tot1 = red[0]; __syncthreads();

    red[tid] = s2; __syncthreads();
#pragma unroll
    for (int s = 128; s > 0; s >>= 1) { if (tid < s) red[tid] += red[tid + s]; __syncthreads(); }
    tot2 = red[0]; __syncthreads();

    red[tid] = s3; __syncthreads();
#pragma unroll
    for (int s = 128; s > 0; s >>= 1) { if (tid < s) red[tid] += red[tid + s]; __syncthreads(); }
    tot3 = red[0];

    if (tid == 0) {
        float mu   = __expf(*p_log_mu);
        float zeta = __expf(*p_log_zeta);
        float term1 = -tot1;
        float term2 = mu * (float)(*p_t_end - *p_t_start) - tot2 * mu;
        float term3 = tot3 / (zeta * zeta * zeta);
        out[0] = term1 + term2 + term3;
    }
}

// ---------------------------------------------------------------------------
extern "C" void kernel_launch(void* const* d_in, const int* in_sizes, int n_in,
                              void* d_out, int out_size, void* d_ws, size_t ws_size,
                              hipStream_t stream) {
    const float* his_t    = (const float*)d_in[0];
    const float* log_mu   = (const float*)d_in[1];
    const float* log_zeta = (const float*)d_in[2];
    const float* c0       = (const float*)d_in[3];
    const float* c1       = (const float*)d_in[4];
    const float* c2       = (const float*)d_in[5];
    const int*   t_start  = (const int*)d_in[6];
    const int*   t_end    = (const int*)d_in[7];

    const int n = in_sizes[0];  // 8192

    float* ws_f    = (float*)d_ws;
    float* ends    = ws_f;                 // n+1
    float* loglamb = ends + (n + 1);       // n
    float* roots   = loglamb + n;          // n
    float* t3p     = roots + n;            // n+1

    hawkes_phaseA<<<(n + 15) / 16, 256, 0, stream>>>(
        his_t, log_mu, log_zeta, c0, c1, c2, n, ends, loglamb, roots);

    hawkes_term3<<<n + 1, 256, 0, stream>>>(
        his_t, log_zeta, c0, c1, c2, t_end, n, ends, t3p);

    hawkes_final<<<1, 256, 0, stream>>>(
        log_mu, log_zeta, t_start, t_end, n, loglamb, roots, t3p, (float*)d_out);
}